// GAPlane_70514773066490
// MI455X (gfx1250) — compile-verified
//
#include <hip/hip_runtime.h>
#include <hip/hip_bf16.h>

// ---------------------------------------------------------------------------
// Fused tri-plane/line/volume field sampler + 64x64 MLP for gfx1250 (MI455X)
//   - prep kernel transposes tables to point-major [cell][64] layout in d_ws
//   - main kernel: phase 1 gathers features (float4 taps, precomputed corner
//                  weights -> mul+fma chains) into LDS
//                  phase 2 runs the 64x64 GEMM with V_WMMA_F32_16X16X4_F32
// ---------------------------------------------------------------------------

typedef __attribute__((ext_vector_type(2))) float v2f;
typedef __attribute__((ext_vector_type(8))) float v8f;

#define FDIM   64
#define RESL   512
#define D2     20
#define D3     5
#define PTS    128          // points per block == threads per block
#define LPAD   68           // LDS row pitch in floats (bank-conflict-free)

#define LINE_T  (3 * RESL * FDIM)        // 98304 floats
#define PLANE_T (3 * D2 * D2 * FDIM)     // 76800 floats
#define VOL_T   (D3 * D3 * D3 * FDIM)    // 8000 floats
#define WS_FLOATS (LINE_T + PLANE_T + VOL_T)

// ---------------------------- helpers --------------------------------------
__device__ __forceinline__ float4 ld4(const float* p) {
    return *reinterpret_cast<const float4*>(p);
}
__device__ __forceinline__ float4 lerp4(float4 a, float4 b, float w) {
    float4 r;
    r.x = fmaf(b.x - a.x, w, a.x);
    r.y = fmaf(b.y - a.y, w, a.y);
    r.z = fmaf(b.z - a.z, w, a.z);
    r.w = fmaf(b.w - a.w, w, a.w);
    return r;
}
__device__ __forceinline__ void grid_prep(float g, float scaleMax, int clampMax,
                                          int& i0, float& w) {
    float pos = (g + 1.0f) * 0.5f * scaleMax;
    pos = fminf(fmaxf(pos, 0.0f), scaleMax);
    int i = (int)floorf(pos);
    i = (i < clampMax) ? i : clampMax;     // exact vs reference for in-range g
    i0 = i;
    w  = pos - (float)i;
}
// bilinear with precomputed corner weights: w = {w00, w01, w10, w11}
// cells: (y0,x0),(y0,x0+1),(y0+1,x0),(y0+1,x0+1); table layout [cell=y*20+x][64]
__device__ __forceinline__ float4 plane4w(const float* base, int x0, int y0,
                                          float4 w, int f) {
    const float* r0 = base + ((y0 * D2 + x0) << 6) + f;
    const float* r1 = r0 + (D2 << 6);
    float4 v00 = ld4(r0),  v01 = ld4(r0 + FDIM);
    float4 v10 = ld4(r1),  v11 = ld4(r1 + FDIM);
    float4 r;
    r.x = fmaf(v11.x, w.w, fmaf(v10.x, w.z, fmaf(v01.x, w.y, v00.x * w.x)));
    r.y = fmaf(v11.y, w.w, fmaf(v10.y, w.z, fmaf(v01.y, w.y, v00.y * w.x)));
    r.z = fmaf(v11.z, w.w, fmaf(v10.z, w.z, fmaf(v01.z, w.y, v00.z * w.x)));
    r.w = fmaf(v11.w, w.w, fmaf(v10.w, w.z, fmaf(v01.w, w.y, v00.w * w.x)));
    return r;
}
// trilinear with 8 precomputed corner weights, order w[dz*4 + dy*2 + dx]
__device__ __forceinline__ float4 volume4w(const float* base, int x0, int y0,
                                           int z0, const float* w, int f) {
    const float* p = base + (((z0 * D3 + y0) * D3 + x0) << 6) + f;
    const float* q = p + ((D3 * D3) << 6);           // z0+1
    float4 v0 = ld4(p),                 v1 = ld4(p + FDIM);
    float4 v2 = ld4(p + (D3 << 6)),     v3 = ld4(p + (D3 << 6) + FDIM);
    float4 v4 = ld4(q),                 v5 = ld4(q + FDIM);
    float4 v6 = ld4(q + (D3 << 6)),     v7 = ld4(q + (D3 << 6) + FDIM);
    float4 r;
    r.x = fmaf(v7.x, w[7], fmaf(v6.x, w[6], fmaf(v5.x, w[5], fmaf(v4.x, w[4],
          fmaf(v3.x, w[3], fmaf(v2.x, w[2], fmaf(v1.x, w[1], v0.x * w[0])))))));
    r.y = fmaf(v7.y, w[7], fmaf(v6.y, w[6], fmaf(v5.y, w[5], fmaf(v4.y, w[4],
          fmaf(v3.y, w[3], fmaf(v2.y, w[2], fmaf(v1.y, w[1], v0.y * w[0])))))));
    r.z = fmaf(v7.z, w[7], fmaf(v6.z, w[6], fmaf(v5.z, w[5], fmaf(v4.z, w[4],
          fmaf(v3.z, w[3], fmaf(v2.z, w[2], fmaf(v1.z, w[1], v0.z * w[0])))))));
    r.w = fmaf(v7.w, w[7], fmaf(v6.w, w[6], fmaf(v5.w, w[5], fmaf(v4.w, w[4],
          fmaf(v3.w, w[3], fmaf(v2.w, w[2], fmaf(v1.w, w[1], v0.w * w[0])))))));
    return r;
}

// -------------------- prep: transpose tables into d_ws ----------------------
__global__ void prep_transpose(const float* __restrict__ lx,
                               const float* __restrict__ ly,
                               const float* __restrict__ lz,
                               const float* __restrict__ pxy,
                               const float* __restrict__ pyz,
                               const float* __restrict__ pzx,
                               const float* __restrict__ vol,
                               float* __restrict__ ws) {
    int idx = blockIdx.x * 256 + threadIdx.x;
    if (idx < LINE_T) {
        int a = idx / (RESL * FDIM);
        int r = idx % (RESL * FDIM);
        int i = r >> 6, f = r & 63;
        const float* src = (a == 0) ? lx : (a == 1) ? ly : lz;
        ws[idx] = src[f * RESL + i];
    } else if (idx < LINE_T + PLANE_T) {
        int t = idx - LINE_T;
        int a = t / (D2 * D2 * FDIM);
        int r = t % (D2 * D2 * FDIM);
        int cell = r >> 6, f = r & 63;
        const float* src = (a == 0) ? pxy : (a == 1) ? pyz : pzx;
        ws[idx] = src[f * (D2 * D2) + cell];
    } else if (idx < WS_FLOATS) {
        int t = idx - LINE_T - PLANE_T;
        int cell = t >> 6, f = t & 63;
        ws[idx] = vol[f * (D3 * D3 * D3) + cell];
    }
}

// ------------------------------ main kernel ---------------------------------
__launch_bounds__(PTS)
__global__ void fused_field_mlp(const float* __restrict__ coords,
                                const float* __restrict__ ws,
                                const float* __restrict__ w1,
                                const float* __restrict__ b1,
                                const float* __restrict__ w2,
                                const float* __restrict__ b2,
                                float* __restrict__ out, int n) {
    __shared__ float sfeat[PTS * LPAD];     // 34,816 B
    __shared__ float sw1[FDIM * LPAD];      // 17,408 B

    const int tid = threadIdx.x;

    // stage W1 into LDS (row = output g, col = input f), padded pitch
    for (int i = tid; i < FDIM * FDIM; i += PTS)
        sw1[(i >> 6) * LPAD + (i & 63)] = w1[i];

    const float* lineTx   = ws;
    const float* lineTy   = ws + RESL * FDIM;
    const float* lineTz   = ws + 2 * RESL * FDIM;
    const float* planeTxy = ws + LINE_T;
    const float* planeTyz = planeTxy + D2 * D2 * FDIM;
    const float* planeTzx = planeTyz + D2 * D2 * FDIM;
    const float* volT     = planeTzx + D2 * D2 * FDIM;

    // ---------------- phase 1: one thread = one point -----------------------
    int pt = blockIdx.x * PTS + tid;
    if (pt >= n) pt = n - 1;                 // safe clamp (n % PTS == 0 here)
    const float cx = coords[3 * pt + 0];
    const float cy = coords[3 * pt + 1];
    const float cz = coords[3 * pt + 2];

    int ixL, iyL, izL;   float wxL, wyL, wzL;
    grid_prep(cx, (float)(RESL - 1), RESL - 2, ixL, wxL);
    grid_prep(cy, (float)(RESL - 1), RESL - 2, iyL, wyL);
    grid_prep(cz, (float)(RESL - 1), RESL - 2, izL, wzL);

    int ixP, iyP, izP;   float wxP, wyP, wzP;
    grid_prep(cx, (float)(D2 - 1), D2 - 2, ixP, wxP);
    grid_prep(cy, (float)(D2 - 1), D2 - 2, iyP, wyP);
    grid_prep(cz, (float)(D2 - 1), D2 - 2, izP, wzP);

    int ixV, iyV, izV;   float wxV, wyV, wzV;
    grid_prep(cx, (float)(D3 - 1), D3 - 2, ixV, wxV);
    grid_prep(cy, (float)(D3 - 1), D3 - 2, iyV, wyV);
    grid_prep(cz, (float)(D3 - 1), D3 - 2, izV, wzV);

    // precomputed bilinear corner weights per plane (w00,w01,w10,w11)
    // plane_xy(x,y): wx=wxP, wy=wyP ; plane_yz(y,z): wx=wyP, wy=wzP ;
    // plane_zx(z,x): wx=wzP, wy=wxP
    const float xP0 = 1.0f - wxP, yP0 = 1.0f - wyP, zP0 = 1.0f - wzP;
    const float4 wXY = { xP0 * yP0, wxP * yP0, xP0 * wyP, wxP * wyP };
    const float4 wYZ = { yP0 * zP0, wyP * zP0, yP0 * wzP, wyP * wzP };
    const float4 wZX = { zP0 * xP0, wzP * xP0, zP0 * wxP, wzP * wxP };

    // precomputed trilinear corner weights, order [dz][dy][dx]
    const float xV0 = 1.0f - wxV, yV0 = 1.0f - wyV, zV0 = 1.0f - wzV;
    float wV[8];
    wV[0] = zV0 * yV0 * xV0;  wV[1] = zV0 * yV0 * wxV;
    wV[2] = zV0 * wyV * xV0;  wV[3] = zV0 * wyV * wxV;
    wV[4] = wzV * yV0 * xV0;  wV[5] = wzV * yV0 * wxV;
    wV[6] = wzV * wyV * xV0;  wV[7] = wzV * wyV * wxV;

    const float* lxr = lineTx + (ixL << 6);
    const float* lyr = lineTy + (iyL << 6);
    const float* lzr = lineTz + (izL << 6);

    float* frow = &sfeat[tid * LPAD];
    #pragma unroll 4
    for (int f = 0; f < FDIM; f += 4) {
        float4 fx = lerp4(ld4(lxr + f), ld4(lxr + FDIM + f), wxL);
        float4 fy = lerp4(ld4(lyr + f), ld4(lyr + FDIM + f), wyL);
        float4 fz = lerp4(ld4(lzr + f), ld4(lzr + FDIM + f), wzL);
        // cell maps: xy -> yc*20+xc ; yz -> zc*20+yc ; zx -> xc*20+zc
        float4 pxy = plane4w(planeTxy, ixP, iyP, wXY, f);
        float4 pyz = plane4w(planeTyz, iyP, izP, wYZ, f);
        float4 pzx = plane4w(planeTzx, izP, ixP, wZX, f);
        float4 vv  = volume4w(volT, ixV, iyV, izV, wV, f);

        // combined = fx*fy*fz + pxy*fz + pyz*fx + pzx*fy + vol
        float4 cmb;
        cmb.x = fmaf(fmaf(fx.x, fy.x, pxy.x), fz.x,
                     fmaf(pyz.x, fx.x, fmaf(pzx.x, fy.x, vv.x)));
        cmb.y = fmaf(fmaf(fx.y, fy.y, pxy.y), fz.y,
                     fmaf(pyz.y, fx.y, fmaf(pzx.y, fy.y, vv.y)));
        cmb.z = fmaf(fmaf(fx.z, fy.z, pxy.z), fz.z,
                     fmaf(pyz.z, fx.z, fmaf(pzx.z, fy.z, vv.z)));
        cmb.w = fmaf(fmaf(fx.w, fy.w, pxy.w), fz.w,
                     fmaf(pyz.w, fx.w, fmaf(pzx.w, fy.w, vv.w)));
        *reinterpret_cast<float4*>(&frow[f]) = cmb;   // 16B aligned (LPAD*4=272)
    }

    __syncthreads();

    // ------------- phase 2: wave32 WMMA tiles (16 points each) --------------
    // f32 WMMA 16x16x4: A lane = row M (lane&15), koff = 2*(lane>>4)
    const int lane = tid & 31;
    const int wv   = tid >> 5;               // wave id 0..3
    const int half = lane >> 4;              // 0 | 1
    const int ln   = lane & 15;
    const int koff = half << 1;
    const float b2v = b2[0];

    for (int tile = wv; tile < PTS / 16; tile += 4) {
        const float* fr = &sfeat[(tile * 16 + ln) * LPAD];
        v8f pacc;
        #pragma unroll
        for (int r = 0; r < 8; ++r) pacc[r] = 0.0f;

        #pragma unroll
        for (int nt = 0; nt < 4; ++nt) {
            const int g0 = nt << 4;
            const float bias = b1[g0 + ln];           // N-dependent bias
            v8f c;
            #pragma unroll
            for (int r = 0; r < 8; ++r) c[r] = bias;

            const float* wr = &sw1[(g0 + ln) * LPAD]; // B[k][n] = w1[g0+n][k]
            #pragma unroll
            for (int k = 0; k < FDIM; k += 4) {
                v2f a, b;
                a.x = fr[k + koff];
                a.y = fr[k + koff + 1];
                b.x = wr[k + koff];
                b.y = wr[k + koff + 1];
                c = __builtin_amdgcn_wmma_f32_16x16x4_f32(
                        false, a, false, b, (short)0, c, false, false);
            }
            // ReLU + layer-2 partial: element (r, lane) = H[M=r+8*half][g0+ln]
            const float w2v = w2[g0 + ln];
            #pragma unroll
            for (int r = 0; r < 8; ++r) {
                float h = c[r] > 0.0f ? c[r] : 0.0f;
                pacc[r] = fmaf(h, w2v, pacc[r]);
            }
        }
        // reduce over N (16 lanes within each half; xor masks < 16 stay in-half)
        #pragma unroll
        for (int s = 1; s < 16; s <<= 1) {
            #pragma unroll
            for (int r = 0; r < 8; ++r)
                pacc[r] += __shfl_xor(pacc[r], s, 32);
        }
        if (ln == 0) {
            const int base = blockIdx.x * PTS + tile * 16 + (half << 3);
            #pragma unroll
            for (int r = 0; r < 8; ++r)
                if (base + r < n) out[base + r] = pacc[r] + b2v;
        }
    }
}

// ------------------------------- host side ----------------------------------
extern "C" void kernel_launch(void* const* d_in, const int* in_sizes, int n_in,
                              void* d_out, int out_size, void* d_ws, size_t ws_size,
                              hipStream_t stream) {
    const float* coords   = (const float*)d_in[0];
    const float* line_x   = (const float*)d_in[1];
    const float* line_y   = (const float*)d_in[2];
    const float* line_z   = (const float*)d_in[3];
    const float* plane_xy = (const float*)d_in[4];
    const float* plane_yz = (const float*)d_in[5];
    const float* plane_zx = (const float*)d_in[6];
    const float* volume   = (const float*)d_in[7];
    const float* w1       = (const float*)d_in[8];
    const float* b1       = (const float*)d_in[9];
    const float* w2       = (const float*)d_in[10];
    const float* b2       = (const float*)d_in[11];
    float* out = (float*)d_out;
    float* ws  = (float*)d_ws;   // needs WS_FLOATS*4 = 732,416 bytes

    const int n = in_sizes[0] / 3;

    const int prepBlocks = (WS_FLOATS + 255) / 256;
    prep_transpose<<<prepBlocks, 256, 0, stream>>>(line_x, line_y, line_z,
                                                   plane_xy, plane_yz, plane_zx,
                                                   volume, ws);

    const int blocks = (n + PTS - 1) / PTS;
    fused_field_mlp<<<blocks, PTS, 0, stream>>>(coords, ws, w1, b1, w2, b2,
                                                out, n);
}